// TMLPCUGO_14027363189340
// MI455X (gfx1250) — compile-verified
//
#include <hip/hip_runtime.h>
#include <math.h>

// ---------------------------------------------------------------------------
// Fused bipartite edge-MLP for MI455X (gfx1250, wave32, WMMA).
//   mlp_src = src_feat @ w_src.T                    [NSRC,128]   (node kernel)
//   mlp_dst = dst_feat @ w_dst.T + b0               [NDST,128]   (node kernel)
//   h = efeat @ w_efeat.T + mlp_src[sidx] + mlp_dst[didx]        (edge kernel)
//   y = LayerNorm(silu(h) @ w_out.T + b_out) * g + b
// One wave32 owns a 16-row tile; GEMMs use V_WMMA_F32_16X16X4_F32 (full f32).
// Streaming traffic (efeat in, y out, node feats in) uses non-temporal cache
// policy so the 102 MB mlp_src/mlp_dst gather tables stay resident in the
// 192 MB L2; gathers then cost L2 bandwidth, not HBM (23.3 TB/s roofline).
// ---------------------------------------------------------------------------

typedef float v2f __attribute__((ext_vector_type(2)));
typedef float v8f __attribute__((ext_vector_type(8)));

#define DD 128
#define K4 32          // 128 / 4 k-steps per GEMM
#define NT 8           // 8 column tiles of 16
#define LDSPITCH 132   // 128 + 4 pad -> conflict-free LDS banking

static __device__ __forceinline__ v8f wmma4(v2f a, v2f b, v8f c) {
  // D = A(16x4) * B(4x16) + C(16x16), all f32
  return __builtin_amdgcn_wmma_f32_16x16x4_f32(false, a, false, b,
                                               (short)0, c, false, false);
}

static __device__ __forceinline__ v2f ld_nt_v2f(const float* p) {
  return __builtin_nontemporal_load((const v2f*)p);   // global_load_b64, TH=NT
}

// ------------------------- node-level linears ------------------------------
__global__ __launch_bounds__(256) void node_linear_kernel(
    const float* __restrict__ src_feat, const float* __restrict__ dst_feat,
    const float* __restrict__ w_src, const float* __restrict__ w_dst,
    const float* __restrict__ b0,
    float* __restrict__ mlp_src, float* __restrict__ mlp_dst,
    int nsrc, int ndst)
{
  const int wave = (blockIdx.x * blockDim.x + threadIdx.x) >> 5;
  const int lane = threadIdx.x & 31;
  const int sTiles = (nsrc + 15) >> 4;
  const int dTiles = (ndst + 15) >> 4;
  if (wave >= sTiles + dTiles) return;

  const float* in; const float* w; const float* bias; float* outp;
  int nrows, r0;
  if (wave < sTiles) {
    in = src_feat; w = w_src; bias = nullptr; outp = mlp_src;
    nrows = nsrc; r0 = wave << 4;
  } else {
    in = dst_feat; w = w_dst; bias = b0; outp = mlp_dst;
    nrows = ndst; r0 = (wave - sTiles) << 4;
  }

  const int l15  = lane & 15;
  const int hi   = lane >> 4;      // 0: K pair (0,1)   1: K pair (2,3)
  const int kofs = hi * 2;
  const int hi8  = hi * 8;

  int arow = r0 + l15; if (arow >= nrows) arow = nrows - 1;
  const float* arowp = in + (size_t)arow * DD;

  v8f acc[NT];
  #pragma unroll
  for (int nt = 0; nt < NT; ++nt) acc[nt] = (v8f)0.0f;

  #pragma unroll
  for (int k = 0; k < K4; ++k) {
    v2f a = ld_nt_v2f(arowp + k * 4 + kofs);  // node feats: streamed once
    #pragma unroll
    for (int nt = 0; nt < NT; ++nt) {
      v2f b = *(const v2f*)(w + (size_t)(nt * 16 + l15) * DD + k * 4 + kofs);
      acc[nt] = wmma4(a, b, acc[nt]);
    }
  }

  // mlp_* tables: DEFAULT (RT) stores -> want them resident in L2 for gathers
  #pragma unroll
  for (int nt = 0; nt < NT; ++nt) {
    const int n = nt * 16 + l15;
    const float bb = bias ? bias[n] : 0.0f;
    #pragma unroll
    for (int i = 0; i < 8; ++i) {
      const int row = r0 + i + hi8;
      if (row < nrows) outp[(size_t)row * DD + n] = acc[nt][i] + bb;
    }
  }
}

// ------------------------- fused edge MLP ----------------------------------
__global__ __launch_bounds__(256) void edge_mlp_kernel(
    const float* __restrict__ efeat,
    const int* __restrict__ src_idx, const int* __restrict__ dst_idx,
    const float* __restrict__ w_efeat, const float* __restrict__ w_out,
    const float* __restrict__ b_out,
    const float* __restrict__ ln_g, const float* __restrict__ ln_b,
    const float* __restrict__ mlp_src, const float* __restrict__ mlp_dst,
    float* __restrict__ out, int E)
{
  __shared__ float lds[8][16 * LDSPITCH];   // 8 waves x 16x132 f32 = 67.6 KB

  const int waveInBlk = threadIdx.x >> 5;
  const int lane = threadIdx.x & 31;
  const int tile = blockIdx.x * (blockDim.x >> 5) + waveInBlk;
  const int tiles = (E + 15) >> 4;
  if (tile >= tiles) return;                 // wave-uniform: EXEC stays all-1s
  const int e0 = tile << 4;

  const int l15  = lane & 15;
  const int hi   = lane >> 4;
  const int kofs = hi * 2;
  const int hi8  = hi * 8;

  int arow = e0 + l15; if (arow >= E) arow = E - 1;
  const float* arowp = efeat + (size_t)arow * DD;

  // ---- GEMM1: h = efeat @ w_efeat.T  (efeat streamed non-temporally) ----
  v8f acc[NT];
  #pragma unroll
  for (int nt = 0; nt < NT; ++nt) acc[nt] = (v8f)0.0f;

  #pragma unroll
  for (int k = 0; k < K4; ++k) {
    v2f a = ld_nt_v2f(arowp + k * 4 + kofs);
    #pragma unroll
    for (int nt = 0; nt < NT; ++nt) {
      v2f b = *(const v2f*)(w_efeat + (size_t)(nt * 16 + l15) * DD + k * 4 + kofs);
      acc[nt] = wmma4(a, b, acc[nt]);
    }
  }

  // ---- gather node terms (L2-resident, RT policy), SiLU, spill to LDS ----
  float* hrow = lds[waveInBlk];              // wave-private slice
  #pragma unroll
  for (int i = 0; i < 8; ++i) {
    const int m = i + hi8;
    int er = e0 + m; if (er >= E) er = E - 1;
    const float* sp = mlp_src + (size_t)src_idx[er] * DD;
    const float* dp = mlp_dst + (size_t)dst_idx[er] * DD;
    #pragma unroll
    for (int nt = 0; nt < NT; ++nt) {
      const int n = nt * 16 + l15;
      float v = acc[nt][i] + sp[n] + dp[n];
      v = v * (1.0f / (1.0f + __expf(-v)));  // SiLU
      hrow[m * LDSPITCH + n] = v;
    }
  }

  // ---- GEMM2: y = silu(h) @ w_out.T  (A frags reloaded from LDS) ----
  #pragma unroll
  for (int nt = 0; nt < NT; ++nt) acc[nt] = (v8f)0.0f;

  #pragma unroll
  for (int k = 0; k < K4; ++k) {
    v2f a = *(const v2f*)(&hrow[l15 * LDSPITCH + k * 4 + kofs]);
    #pragma unroll
    for (int nt = 0; nt < NT; ++nt) {
      v2f b = *(const v2f*)(w_out + (size_t)(nt * 16 + l15) * DD + k * 4 + kofs);
      acc[nt] = wmma4(a, b, acc[nt]);
    }
  }

  // ---- bias + LayerNorm + store ----
  float bo[NT], gg[NT], bb[NT];
  #pragma unroll
  for (int nt = 0; nt < NT; ++nt) {
    const int n = nt * 16 + l15;
    bo[nt] = b_out[n]; gg[nt] = ln_g[n]; bb[nt] = ln_b[n];
  }

  #pragma unroll
  for (int i = 0; i < 8; ++i) {
    // row m = i + hi*8 lives across one 16-lane half; reduce within the half
    float s1 = 0.0f;
    #pragma unroll
    for (int nt = 0; nt < NT; ++nt) s1 += acc[nt][i] + bo[nt];
    s1 += __shfl_xor(s1, 1); s1 += __shfl_xor(s1, 2);
    s1 += __shfl_xor(s1, 4); s1 += __shfl_xor(s1, 8);
    const float mu = s1 * (1.0f / 128.0f);

    float s2 = 0.0f;
    #pragma unroll
    for (int nt = 0; nt < NT; ++nt) {
      const float d = acc[nt][i] + bo[nt] - mu;
      s2 += d * d;
    }
    s2 += __shfl_xor(s2, 1); s2 += __shfl_xor(s2, 2);
    s2 += __shfl_xor(s2, 4); s2 += __shfl_xor(s2, 8);
    const float rstd = rsqrtf(s2 * (1.0f / 128.0f) + 1e-5f);

    const int er = e0 + i + hi8;
    if (er < E) {
      float* op = out + (size_t)er * DD;
      #pragma unroll
      for (int nt = 0; nt < NT; ++nt) {
        const float d = acc[nt][i] + bo[nt] - mu;
        // output streamed non-temporally: don't evict gather tables from L2
        __builtin_nontemporal_store(d * rstd * gg[nt] + bb[nt],
                                    op + nt * 16 + l15);
      }
    }
  }
}

// ------------------------- host launch -------------------------------------
extern "C" void kernel_launch(void* const* d_in, const int* in_sizes, int n_in,
                              void* d_out, int out_size, void* d_ws, size_t ws_size,
                              hipStream_t stream) {
  const float* efeat    = (const float*)d_in[0];
  const float* src_feat = (const float*)d_in[1];
  const float* dst_feat = (const float*)d_in[2];
  const int*   src_idx  = (const int*)d_in[3];
  const int*   dst_idx  = (const int*)d_in[4];
  const float* w_efeat  = (const float*)d_in[5];
  const float* w_src    = (const float*)d_in[6];
  const float* w_dst    = (const float*)d_in[7];
  const float* b0       = (const float*)d_in[8];
  const float* w_out    = (const float*)d_in[9];
  const float* b_out    = (const float*)d_in[10];
  const float* ln_g     = (const float*)d_in[11];
  const float* ln_b     = (const float*)d_in[12];
  (void)n_in; (void)out_size; (void)ws_size;

  const int E    = in_sizes[0] / DD;
  const int NSRC = in_sizes[1] / DD;
  const int NDST = in_sizes[2] / DD;

  float* mlp_src = (float*)d_ws;                       // [NSRC,128] f32
  float* mlp_dst = mlp_src + (size_t)NSRC * DD;        // [NDST,128] f32

  {
    const int tiles  = ((NSRC + 15) >> 4) + ((NDST + 15) >> 4);
    const int blocks = (tiles + 7) / 8;                // 8 waves per block
    node_linear_kernel<<<blocks, 256, 0, stream>>>(
        src_feat, dst_feat, w_src, w_dst, b0, mlp_src, mlp_dst, NSRC, NDST);
  }
  {
    const int tiles  = (E + 15) >> 4;
    const int blocks = (tiles + 7) / 8;
    edge_mlp_kernel<<<blocks, 256, 0, stream>>>(
        efeat, src_idx, dst_idx, w_efeat, w_out, b_out, ln_g, ln_b,
        mlp_src, mlp_dst, (float*)d_out, E);
  }
}